// RoPE_43791486550551
// MI455X (gfx1250) — compile-verified
//
#include <hip/hip_runtime.h>

// RoPE forward: x (2,16,4096,128) f32 -> out same shape.
// out[b,h,s,2k]   = c*x[2k] - s*x[2k+1]
// out[b,h,s,2k+1] = s*x[2k] + c*x[2k+1],  c,s = cos/sin(pos * theta^(-2k/128))
//
// Pure HBM-streaming kernel (134 MB total -> ~5.8us @ 23.3 TB/s). Coefficients
// are recomputed with TRANS ops (v_exp/v_sin/v_cos) instead of gathering them
// out of the 256MB mostly-zero rot_mats (1032B-stride 4B gather = ~64MB of
// wasted line traffic). x is staged through LDS with gfx1250 async
// global->LDS B128 copies (ASYNCcnt-tracked, depth-2 pipeline), NT cache
// policy on both the streaming loads and stores.

#define WAVES_PER_BLOCK 8
#define ROWS_PER_WAVE   8          // 32 bh rows / gridDim.y(4)
#define SEQ             4096
#define DMODEL          128
#define ROW_ELEMS       (SEQ * DMODEL)          // elements between (b,h) slices
#define ROW_BYTES       (ROW_ELEMS * 4u)        // 2 MiB

typedef float v4f __attribute__((ext_vector_type(4)));

// GVS-form async copy: SGPR 64-bit base + 32-bit VGPR byte offset (saves the
// per-row 64-bit VALU address adds of the GV form). Non-temporal: data is
// touched exactly once.
__device__ __forceinline__ void async_row_b128(unsigned lds_byte_addr,
                                               const float* uniform_base,
                                               unsigned voff_bytes) {
    asm volatile("global_load_async_to_lds_b128 %0, %1, %2 th:TH_LOAD_NT"
                 :
                 : "v"(lds_byte_addr),
                   "v"(voff_bytes),
                   "s"((unsigned long long)(const void*)uniform_base)
                 : "memory");
}

__global__ __launch_bounds__(256) void rope_fwd_kernel(
    const float* __restrict__ x,
    const int*   __restrict__ token_positions,
    float*       __restrict__ out)
{
    __shared__ __align__(16) float lbuf[WAVES_PER_BLOCK][2][DMODEL];

    const int tid  = threadIdx.x;
    const int wave = tid >> 5;
    const int lane = tid & 31;

    const int s   = blockIdx.x * WAVES_PER_BLOCK + wave;   // sequence position
    const int bh0 = blockIdx.y * ROWS_PER_WAVE;            // first (b*h) row

    // Per-lane coefficients for pairs k0 = 2*lane, k1 = 2*lane+1; computed
    // once per wave, reused across all 8 rows.
    // theta^(-2k/128) = exp(-(ln 1e4)/64 * k);  ln(10000)/64 = 0.14391157
    const float posf = (float)token_positions[s];
    const float k0   = (float)(2 * lane);
    const float a0   = posf * __expf(-0.14391157f * k0);
    const float a1   = posf * __expf(-0.14391157f * (k0 + 1.0f));
    const float c0 = __cosf(a0), s0 = __sinf(a0);
    const float c1 = __cosf(a1), s1 = __sinf(a1);

    // 32-bit addressing: tensor is 64 MiB, every offset fits in u32.
    const unsigned base_e = ((unsigned)bh0 * SEQ + (unsigned)s) * DMODEL
                          + (unsigned)lane * 4u;            // element index
    const unsigned base_b = base_e * 4u;                    // byte offset

    const unsigned ldsA = (unsigned)(size_t)&lbuf[wave][0][lane * 4];
    const unsigned ldsB = (unsigned)(size_t)&lbuf[wave][1][lane * 4];

    // Depth-2 software pipeline on ASYNCcnt: row r+1 in flight while row r
    // is rotated and stored.
    async_row_b128(ldsA, x, base_b);
#pragma unroll
    for (int r = 0; r < ROWS_PER_WAVE; ++r) {
        if (r + 1 < ROWS_PER_WAVE) {
            async_row_b128(((r + 1) & 1) ? ldsB : ldsA,
                           x, base_b + (unsigned)(r + 1) * ROW_BYTES);
            asm volatile("s_wait_asynccnt 1" ::: "memory");  // oldest copy done
        } else {
            asm volatile("s_wait_asynccnt 0" ::: "memory");
        }

        const v4f xv = *reinterpret_cast<const v4f*>(
            (r & 1) ? &lbuf[wave][1][lane * 4] : &lbuf[wave][0][lane * 4]);

        v4f y;
        y.x = fmaf(c0, xv.x, -s0 * xv.y);   // even = c*xe - s*xo
        y.y = fmaf(s0, xv.x,  c0 * xv.y);   // odd  = s*xe + c*xo
        y.z = fmaf(c1, xv.z, -s1 * xv.w);
        y.w = fmaf(s1, xv.z,  c1 * xv.w);

        // Streaming store, written once: NT policy, saddr+u32-offset form.
        __builtin_nontemporal_store(
            y, reinterpret_cast<v4f*>(out + base_e + (unsigned)r * ROW_ELEMS));
    }
}

extern "C" void kernel_launch(void* const* d_in, const int* in_sizes, int n_in,
                              void* d_out, int out_size, void* d_ws, size_t ws_size,
                              hipStream_t stream) {
    (void)in_sizes; (void)n_in; (void)out_size; (void)d_ws; (void)ws_size;

    const float* x  = (const float*)d_in[0];
    const int*   tp = (const int*)d_in[1];    // jnp default: int32 (x64 disabled)
    // d_in[2] = rot_mats: intentionally unused (see header comment).
    float* out = (float*)d_out;

    dim3 block(32 * WAVES_PER_BLOCK);                      // 8 waves (wave32)
    dim3 grid(SEQ / WAVES_PER_BLOCK, 32 / ROWS_PER_WAVE);  // (512, 4)
    rope_fwd_kernel<<<grid, block, 0, stream>>>(x, tp, out);
}